// CompressiveMemory_53987738911174
// MI455X (gfx1250) — compile-verified
//
#include <hip/hip_runtime.h>

// CDNA5 / gfx1250 wave32 WMMA types
typedef __attribute__((ext_vector_type(2))) float v2f;
typedef __attribute__((ext_vector_type(4))) float v4f;
typedef __attribute__((ext_vector_type(8))) float v8f;

#define EPS_F 1e-6f
#define Bb 2
#define Hh 32
#define Tt 4096
#define Dd 128
#define SPLITK 16                 // update split-K chunks (256 t-rows each)

// phi(x) = elu(x) + 1 = x+1 (x>0) else exp(x)
__device__ __forceinline__ float phi_f(float x) {
    return x > 0.0f ? (x + 1.0f) : __expf(x);
}
__device__ __forceinline__ v4f phi_v4(v4f x) {
    v4f r;
    r.x = phi_f(x.x); r.y = phi_f(x.y); r.z = phi_f(x.z); r.w = phi_f(x.w);
    return r;
}

// ---------------------------------------------------------------------------
// Kernel 1: out[b,h,t,:] = (phi(Q[b,h,t,:]) @ M[h]) / max(phi(Q[b,h,t,:])·z[h], EPS)
// One block = 64 rows of one (b,h). 8 waves, wave w owns N-strip [16w,16w+16).
// LDS packed as v2f over K: each WMMA fragment = one ds_load_b64.
// Q streamed non-temporal; M kept cacheable (reused by 128 blocks/head).
// ---------------------------------------------------------------------------
__global__ void __launch_bounds__(256)
retrieve_kernel(const float* __restrict__ Q, const float* __restrict__ M,
                const float* __restrict__ z, float* __restrict__ out)
{
    // ldsM2[kp][e]  = { M[2kp][e],    M[2kp+1][e] }    row stride 144 v2f (288 fl ≡ 32 banks)
    // ldsQt2[dp][t] = { phiQ[t][2dp], phiQ[t][2dp+1] } row stride  80 v2f (160 fl ≡ 32 banks)
    __shared__ v2f   ldsM2[64 * 144];   // 73,728 B
    __shared__ v2f   ldsQt2[64 * 80];   // 40,960 B
    __shared__ float ldsZ[Dd];
    __shared__ float ldsNorm[64];

    const int tid    = threadIdx.x;
    const int nblkT  = Tt / 64;
    const int bh     = blockIdx.x / nblkT;
    const int rowblk = blockIdx.x % nblkT;
    const int h      = bh % Hh;

    const float* Mh = M + (size_t)h * Dd * Dd;
    const float* zh = z + (size_t)h * Dd;
    const float* Qb = Q + (size_t)bh * Tt * Dd + (size_t)rowblk * 64 * Dd;
    float* outb     = out + (size_t)bh * Tt * Dd + (size_t)rowblk * 64 * Dd;

    float* ldsMf = (float*)ldsM2;
    float* ldsQf = (float*)ldsQt2;

    // stage M[h]: rows {2kp,2kp+1} x 4 e's -> interleave -> 2x ds_store_b128
    const v4f* M4 = (const v4f*)Mh;
    #pragma unroll
    for (int i = 0; i < 8; ++i) {
        int g  = tid + 256 * i;      // 0..2047 groups of 8 floats
        int kp = g >> 5;             // 0..63
        int eq = g & 31;             // e = 4*eq
        v4f r0 = M4[(2 * kp)     * 32 + eq];
        v4f r1 = M4[(2 * kp + 1) * 32 + eq];
        v4f s0 = {r0.x, r1.x, r0.y, r1.y};
        v4f s1 = {r0.z, r1.z, r0.w, r1.w};
        *(v4f*)&ldsMf[kp * 288 + 8 * eq]     = s0;
        *(v4f*)&ldsMf[kp * 288 + 8 * eq + 4] = s1;
    }
    if (tid < Dd) ldsZ[tid] = zh[tid];

    // stage phi(Q) transposed+paired: NT b128 global load -> 2x ds_store_b64
    const v4f* Q4 = (const v4f*)Qb;
    #pragma unroll
    for (int i = 0; i < 8; ++i) {
        int g  = tid + 256 * i;      // 0..2047
        int t  = g >> 5;             // 0..63
        int dq = g & 31;             // d = 4*dq, dp = 2*dq
        v4f q  = phi_v4(__builtin_nontemporal_load(&Q4[t * 32 + dq]));
        int dp = 2 * dq;
        v2f s0 = {q.x, q.y};
        v2f s1 = {q.z, q.w};
        *(v2f*)&ldsQf[dp * 160 + 2 * t]       = s0;
        *(v2f*)&ldsQf[(dp + 1) * 160 + 2 * t] = s1;
    }
    __syncthreads();

    // per-row denominator: max(phi(q_t)·z, EPS)  (b64 LDS reads, conflict-free)
    if (tid < 64) {
        float s = 0.0f;
        #pragma unroll 8
        for (int dp = 0; dp < 64; ++dp) {
            v2f q = ldsQt2[dp * 80 + tid];
            s += q.x * ldsZ[2 * dp] + q.y * ldsZ[2 * dp + 1];
        }
        ldsNorm[tid] = fmaxf(s, EPS_F);
    }
    __syncthreads();

    const int wave = tid >> 5;
    const int lane = tid & 31;
    const int lo   = lane & 15;
    const int hi   = lane >> 4;
    const int e0   = wave * 16;          // N-strip

    for (int rt = 0; rt < 4; ++rt) {     // 4 row tiles of 16
        const int m0 = rt * 16;
        v8f acc = {};
        #pragma unroll
        for (int k0 = 0; k0 < Dd; k0 += 4) {   // K sweep, 4 per WMMA
            const int kp = (k0 >> 1) + hi;     // pair index (ka = 2*kp)
            v2f a = ldsQt2[kp * 80  + m0 + lo];  // {phiQ[m][ka], phiQ[m][ka+1]}
            v2f b = ldsM2 [kp * 144 + e0 + lo];  // {M[ka][n],    M[ka+1][n]}
            acc = __builtin_amdgcn_wmma_f32_16x16x4_f32(
                    false, a, false, b, (short)0, acc, false, false);
        }
        // C layout: vgpr r, lanes0-15 -> row m0+r, lanes16-31 -> row m0+r+8; col = e0+lo
        #pragma unroll
        for (int r = 0; r < 8; ++r) {
            const int row = m0 + r + 8 * hi;
            __builtin_nontemporal_store(acc[r] / ldsNorm[row],
                                        &outb[row * Dd + e0 + lo]);
        }
    }
}

// ---------------------------------------------------------------------------
// Kernel 2 (init): outM = M, outZ = z  (so split-K atomics land on M + sum)
// ---------------------------------------------------------------------------
__global__ void __launch_bounds__(256)
init_update_out(const float* __restrict__ M, const float* __restrict__ z,
                float* __restrict__ outM, float* __restrict__ outZ)
{
    const int i = blockIdx.x * 256 + threadIdx.x;
    if (i < Hh * Dd * Dd) outM[i] = M[i];
    if (i < Hh * Dd)      outZ[i] = z[i];
}

// ---------------------------------------------------------------------------
// Kernel 3: M_new[h] += phiK_mean[h]^T @ V_mean[h]; z_new[h] += sum_t phiK_mean
// Grid: H * 16 split-K chunks (256 t-rows each). Wave w owns D-rows [16w,16w+16)
// against all 8 E-tiles. Software-pipelined double-buffered LDS: next stage's
// NT global loads issue before the WMMA burst, stores land in the alternate
// buffer after it; one barrier per stage.
// ---------------------------------------------------------------------------
__global__ void __launch_bounds__(256)
update_kernel(const float* __restrict__ K, const float* __restrict__ V,
              float* __restrict__ outM, float* __restrict__ outZ)
{
    // ldsA2[buf][tp][d] = { phiK_mean[2tp][d], phiK_mean[2tp+1][d] }, stride 144 v2f
    __shared__ v2f ldsA2[2][16 * 144];   // 2 x 18,432 B
    __shared__ v2f ldsB2[2][16 * 144];   // 2 x 18,432 B

    const int tid   = threadIdx.x;
    const int h     = blockIdx.x / SPLITK;
    const int chunk = blockIdx.x % SPLITK;
    const int t0    = chunk * (Tt / SPLITK);
    const int NST   = (Tt / SPLITK) / 32;     // 8 stages of 32 rows

    const v4f* K04 = (const v4f*)(K + ((size_t)(0 * Hh + h) * Tt + t0) * Dd);
    const v4f* K14 = (const v4f*)(K + ((size_t)(1 * Hh + h) * Tt + t0) * Dd);
    const v4f* V04 = (const v4f*)(V + ((size_t)(0 * Hh + h) * Tt + t0) * Dd);
    const v4f* V14 = (const v4f*)(V + ((size_t)(1 * Hh + h) * Tt + t0) * Dd);

    const int wave = tid >> 5;
    const int lane = tid & 31;
    const int lo   = lane & 15;
    const int hi   = lane >> 4;
    const int m0   = wave * 16;        // D-row strip

    // staging registers: 2 groups x (A,B) x 2 quads
    v4f rA0[2], rA1[2], rB0[2], rB1[2];

    auto load_stage = [&](int ts) {
        #pragma unroll
        for (int i = 0; i < 2; ++i) {
            int g  = tid + 256 * i;    // 0..511
            int tp = g >> 5;           // 0..15
            int dq = g & 31;           // d = 4*dq
            int t  = ts + 2 * tp;
            v4f pa = phi_v4(__builtin_nontemporal_load(&K04[(size_t)t * 32 + dq]));
            v4f pb = phi_v4(__builtin_nontemporal_load(&K04[(size_t)(t + 1) * 32 + dq]));
            v4f qa = phi_v4(__builtin_nontemporal_load(&K14[(size_t)t * 32 + dq]));
            v4f qb = phi_v4(__builtin_nontemporal_load(&K14[(size_t)(t + 1) * 32 + dq]));
            pa = 0.5f * (pa + qa);
            pb = 0.5f * (pb + qb);
            rA0[i] = (v4f){pa.x, pb.x, pa.y, pb.y};
            rA1[i] = (v4f){pa.z, pb.z, pa.w, pb.w};
            v4f va = 0.5f * (__builtin_nontemporal_load(&V04[(size_t)t * 32 + dq]) +
                             __builtin_nontemporal_load(&V14[(size_t)t * 32 + dq]));
            v4f vb = 0.5f * (__builtin_nontemporal_load(&V04[(size_t)(t + 1) * 32 + dq]) +
                             __builtin_nontemporal_load(&V14[(size_t)(t + 1) * 32 + dq]));
            rB0[i] = (v4f){va.x, vb.x, va.y, vb.y};
            rB1[i] = (v4f){va.z, vb.z, va.w, vb.w};
        }
    };
    auto store_stage = [&](int buf) {
        float* af = (float*)ldsA2[buf];
        float* bf = (float*)ldsB2[buf];
        #pragma unroll
        for (int i = 0; i < 2; ++i) {
            int g  = tid + 256 * i;
            int tp = g >> 5;
            int dq = g & 31;
            *(v4f*)&af[tp * 288 + 8 * dq]     = rA0[i];
            *(v4f*)&af[tp * 288 + 8 * dq + 4] = rA1[i];
            *(v4f*)&bf[tp * 288 + 8 * dq]     = rB0[i];
            *(v4f*)&bf[tp * 288 + 8 * dq + 4] = rB1[i];
        }
    };

    v8f acc[8] = {};
    float zacc = 0.0f;

    load_stage(0);
    store_stage(0);

    for (int s = 0; s < NST; ++s) {
        __syncthreads();               // buf[s&1] visible; buf[(s+1)&1] free
        const int buf = s & 1;
        if (s + 1 < NST) load_stage(32 * (s + 1));   // NT loads overlap WMMA burst

        // z partial: one thread per d; each v2f holds two consecutive t's at d
        if (tid < Dd) {
            float zs = 0.0f;
            #pragma unroll
            for (int tp = 0; tp < 16; ++tp) {
                v2f a = ldsA2[buf][tp * 144 + tid];
                zs += a.x + a.y;
            }
            zacc += zs;
        }

        #pragma unroll
        for (int k0 = 0; k0 < 32; k0 += 4) {
            const int kp = (k0 >> 1) + hi;
            v2f a = ldsA2[buf][kp * 144 + m0 + lo];       // {phiK[ka][m], phiK[ka+1][m]}
            #pragma unroll
            for (int e = 0; e < 8; ++e) {
                v2f b = ldsB2[buf][kp * 144 + e * 16 + lo]; // {V[ka][n], V[ka+1][n]}
                acc[e] = __builtin_amdgcn_wmma_f32_16x16x4_f32(
                           false, a, false, b, (short)0, acc[e], false, false);
            }
        }

        if (s + 1 < NST) store_stage((s + 1) & 1);   // safe: last readers synced above
    }

    float* Mh = outM + (size_t)h * Dd * Dd;
    #pragma unroll
    for (int e = 0; e < 8; ++e) {
        #pragma unroll
        for (int r = 0; r < 8; ++r) {
            const int row = m0 + r + 8 * hi;
            atomicAdd(&Mh[row * Dd + e * 16 + lo], acc[e][r]);
        }
    }
    if (tid < Dd) atomicAdd(&outZ[(size_t)h * Dd + tid], zacc);
}

// ---------------------------------------------------------------------------
extern "C" void kernel_launch(void* const* d_in, const int* in_sizes, int n_in,
                              void* d_out, int out_size, void* d_ws, size_t ws_size,
                              hipStream_t stream)
{
    (void)in_sizes; (void)n_in; (void)out_size; (void)d_ws; (void)ws_size;
    const float* Q = (const float*)d_in[0];
    const float* K = (const float*)d_in[1];
    const float* V = (const float*)d_in[2];
    const float* M = (const float*)d_in[3];
    const float* z = (const float*)d_in[4];

    float* out  = (float*)d_out;
    float* outM = out  + (size_t)Bb * Hh * Tt * Dd;   // 33,554,432
    float* outZ = outM + (size_t)Hh * Dd * Dd;        // +524,288

    // outM/outZ = M/z first (stream-ordered before atomics -> deterministic value)
    init_update_out<<<(Hh * Dd * Dd + 255) / 256, 256, 0, stream>>>(M, z, outM, outZ);
    // retrieve: 2*32*64 = 4096 blocks
    retrieve_kernel<<<Bb * Hh * (Tt / 64), 256, 0, stream>>>(Q, M, z, out);
    // update: 32 heads * 16 split-K chunks
    update_kernel<<<Hh * SPLITK, 256, 0, stream>>>(K, V, outM, outZ);
}